// HeteroGraphSAGE_712964571450
// MI455X (gfx1250) — compile-verified
//
#include <hip/hip_runtime.h>
#include <math.h>

#define NC 50000
#define NM 20000
#define ND 5000
#define DIM 128

#define GEMM_ROWS 64          // rows per block (4 row-tiles of 16 per wave)
#define GEMM_KC   128         // K-chunk staged in LDS
#define LDS_STRIDE (GEMM_KC + 4)  // pad: stride 132 ≡ 4 (mod 64 banks) -> conflict-free

typedef __attribute__((ext_vector_type(2)))  float  v2f;
typedef __attribute__((ext_vector_type(8)))  float  v8f;

// ---------------------------------------------------------------------------
// Edge-parallel scatter-add: one wave per edge, 4 floats/lane (128 feats).
// Accumulates x_src[src] into Xaug[dst, colbase:colbase+128] and counts.
// ---------------------------------------------------------------------------
__global__ void scatter_accum_kernel(const float* __restrict__ xsrc,
                                     const int*   __restrict__ src,
                                     const int*   __restrict__ dst,
                                     float* __restrict__ Xaug,
                                     float* __restrict__ cnt,
                                     int n_edges, int K, int colbase)
{
    int gw   = (int)((blockIdx.x * blockDim.x + threadIdx.x) >> 5);
    int lane = threadIdx.x & 31;
    int nw   = (int)((gridDim.x * blockDim.x) >> 5);
    for (int e = gw; e < n_edges; e += nw) {
        int s = src[e];
        int d = dst[e];
        float4 v = ((const float4*)(xsrc + (size_t)s * DIM))[lane];
        float* base = Xaug + (size_t)d * K + colbase + lane * 4;
        atomicAdd(base + 0, v.x);
        atomicAdd(base + 1, v.y);
        atomicAdd(base + 2, v.z);
        atomicAdd(base + 3, v.w);
        if (lane == 0) atomicAdd(cnt + d, 1.0f);
    }
}

// ---------------------------------------------------------------------------
// Divide accumulated sums by max(cnt,1) per relation slot; append x_dst as the
// final 128 columns of the augmented matrix.
// ---------------------------------------------------------------------------
__global__ void finalize_xaug_kernel(float* __restrict__ Xaug,
                                     const float* __restrict__ cnt,
                                     const float* __restrict__ xdst,
                                     int n, int K, int nrel)
{
    int idx = blockIdx.x * blockDim.x + threadIdx.x;
    if (idx >= n * DIM) return;
    int node = idx >> 7;
    int k    = idx & 127;
    float* row = Xaug + (size_t)node * K;
    for (int s = 0; s < nrel; ++s) {
        float cn = cnt[(size_t)s * n + node];
        row[s * DIM + k] = row[s * DIM + k] / fmaxf(cn, 1.0f);
    }
    row[nrel * DIM + k] = xdst[(size_t)node * DIM + k];
}

// ---------------------------------------------------------------------------
// Build combined weight [128 x K]: Wcomb[j, s*128+k] = Wl[layer,r_s][j,k],
// last 128 cols = sum_s Wr[layer,r_s][j,k]. bsum[j] = sum_s bl[layer,r_s][j].
// ---------------------------------------------------------------------------
__global__ void build_wcomb_kernel(const float* __restrict__ Wl,
                                   const float* __restrict__ Wr,
                                   const float* __restrict__ bl,
                                   float* __restrict__ Wcomb,
                                   float* __restrict__ bsum,
                                   int layer, int r0, int r1, int r2,
                                   int nrel, int K)
{
    int idx = blockIdx.x * blockDim.x + threadIdx.x;
    if (idx >= 128 * 128) return;
    int j = idx >> 7;
    int k = idx & 127;
    int rr[3] = {r0, r1, r2};
    float wr = 0.0f;
    float bs = 0.0f;
    for (int s = 0; s < nrel; ++s) {
        size_t base = ((size_t)layer * 8 + rr[s]) * (size_t)(128 * 128);
        Wcomb[(size_t)j * K + s * DIM + k] = Wl[base + (size_t)j * DIM + k];
        wr += Wr[base + (size_t)j * DIM + k];
        bs += bl[((size_t)layer * 8 + rr[s]) * DIM + j];
    }
    Wcomb[(size_t)j * K + nrel * DIM + k] = wr;
    if (k == 0) bsum[j] = bs;
}

// ---------------------------------------------------------------------------
// WMMA GEMM: out[n,128] = relu? (Xaug[n,K] . Wcomb[128,K]^T + bsum) * scale
// Block = 256 threads = 8 waves. Block owns 64 rows; wave w owns columns
// [16w,16w+16) and 4 row-tiles, so each B fragment load feeds 4 WMMAs.
// A is staged per 128-wide K-chunk in padded LDS (conflict-free ds_load_b64).
// ---------------------------------------------------------------------------
__global__ __launch_bounds__(256) void gemm_wmma_kernel(
    const float* __restrict__ Xaug, const float* __restrict__ W,
    const float* __restrict__ bsum, float* __restrict__ out,
    int n, int K, float scale, int do_relu)
{
    __shared__ float ldsA[GEMM_ROWS * LDS_STRIDE];
    const int tid   = threadIdx.x;
    const int row0  = blockIdx.x * GEMM_ROWS;
    const int lane  = tid & 31;
    const int wv    = tid >> 5;            // 0..7 -> column tile
    const int m16   = lane & 15;
    const int g     = lane >> 4;           // lane half
    const int jt    = wv * 16;
    const int khalf = g * 2;               // A/B frag: lanes 0-15 K{0,1}, 16-31 K{2,3}

    v8f acc[4];
    #pragma unroll
    for (int t = 0; t < 4; ++t)
        acc[t] = (v8f){0.f, 0.f, 0.f, 0.f, 0.f, 0.f, 0.f, 0.f};

    const float* wrow = W + (size_t)(jt + m16) * K;

    for (int kb = 0; kb < K; kb += GEMM_KC) {
        __syncthreads();
        // cooperative float4 load of 64 x 128 A chunk (zero-pad rows >= n)
        for (int idx = tid; idx < (GEMM_ROWS * GEMM_KC / 4); idx += 256) {
            int r  = idx >> 5;             // 32 float4 per row
            int k4 = (idx & 31) * 4;
            int gr = row0 + r;
            float4 v = make_float4(0.f, 0.f, 0.f, 0.f);
            if (gr < n) v = *(const float4*)(Xaug + (size_t)gr * K + kb + k4);
            *(float4*)(ldsA + r * LDS_STRIDE + k4) = v;
        }
        __syncthreads();

        #pragma unroll 4
        for (int k0 = 0; k0 < GEMM_KC; k0 += 4) {
            v2f b;
            b.x = wrow[kb + k0 + khalf];
            b.y = wrow[kb + k0 + khalf + 1];
            #pragma unroll
            for (int t = 0; t < 4; ++t) {
                v2f a;
                a.x = ldsA[(t * 16 + m16) * LDS_STRIDE + k0 + khalf];
                a.y = ldsA[(t * 16 + m16) * LDS_STRIDE + k0 + khalf + 1];
#if __has_builtin(__builtin_amdgcn_wmma_f32_16x16x4_f32)
                acc[t] = __builtin_amdgcn_wmma_f32_16x16x4_f32(
                    false, a, false, b, (short)0, acc[t], false, false);
#else
                // host-pass parse fallback (never executed on gfx1250 device)
                acc[t][0] += a.x * b.x + a.y * b.y;
#endif
            }
        }
    }

    const int j  = jt + m16;
    const float bj = bsum[j];
    #pragma unroll
    for (int t = 0; t < 4; ++t) {
        #pragma unroll
        for (int i = 0; i < 8; ++i) {
            int gr = row0 + t * 16 + g * 8 + i;   // D layout: VGPR i -> row i + 8*half
            if (gr < n) {
                float v = (acc[t][i] + bj) * scale;
                if (do_relu) v = fmaxf(v, 0.0f);
                out[(size_t)gr * DIM + j] = v;
            }
        }
    }
}

// ---------------------------------------------------------------------------
// Row-wise L2 normalize: one wave per row, 4 floats/lane.
// ---------------------------------------------------------------------------
__global__ void l2norm_kernel(const float* __restrict__ in,
                              float* __restrict__ out, int n)
{
    int gw   = (int)((blockIdx.x * blockDim.x + threadIdx.x) >> 5);
    int lane = threadIdx.x & 31;
    int nw   = (int)((gridDim.x * blockDim.x) >> 5);
    for (int row = gw; row < n; row += nw) {
        float4 v = ((const float4*)(in + (size_t)row * DIM))[lane];
        float ss = v.x * v.x + v.y * v.y + v.z * v.z + v.w * v.w;
        for (int o = 16; o > 0; o >>= 1) ss += __shfl_down(ss, o, 32);
        ss = __shfl(ss, 0, 32);
        float inv = 1.0f / fmaxf(sqrtf(ss), 1e-12f);
        float4 r = make_float4(v.x * inv, v.y * inv, v.z * inv, v.w * inv);
        ((float4*)(out + (size_t)row * DIM))[lane] = r;
    }
}

// ---------------------------------------------------------------------------
extern "C" void kernel_launch(void* const* d_in, const int* in_sizes, int n_in,
                              void* d_out, int out_size, void* d_ws, size_t ws_size,
                              hipStream_t stream)
{
    const float* x_c = (const float*)d_in[0];
    const float* x_m = (const float*)d_in[1];
    const float* x_d = (const float*)d_in[2];
    const float* Wl  = (const float*)d_in[3];
    const float* bl  = (const float*)d_in[4];
    const float* Wr  = (const float*)d_in[5];
    const int* E[8]; int ne[8];
    for (int i = 0; i < 8; ++i) {
        E[i]  = (const int*)d_in[6 + i];
        ne[i] = in_sizes[6 + i] / 2;       // [2, n_edges] stacked src;dst
    }

    // Workspace carve-up (floats). Xaug reused across types/layers.
    float* ws = (float*)d_ws;
    size_t off = 0;
    float* Xaug  = ws + off; off += (size_t)NC * 512;   // max n*K
    float* cnt   = ws + off; off += 3 * (size_t)NC;
    float* Wcomb = ws + off; off += (size_t)128 * 512;
    float* bsum  = ws + off; off += 128;
    const int nn[3] = {NC, NM, ND};
    float* hbuf[2][3];
    for (int b = 0; b < 2; ++b)
        for (int t = 0; t < 3; ++t) { hbuf[b][t] = ws + off; off += (size_t)nn[t] * DIM; }

    // Per-dst-type relation tables. RELS order:
    // 0:c->m 1:m->d 2:c->d 3:m->c 4:d->m 5:d->c 6:c->c 7:d->d
    const int nrel[3]       = {3, 2, 3};                 // t: 0=c, 1=m, 2=d
    const int rel_of[3][3]  = { {3, 5, 6}, {0, 4, -1}, {1, 2, 7} };
    const int srct_of[3][3] = { {1, 2, 0}, {0, 2, -1}, {1, 0, 2} };
    const int Kt[3]         = {512, 384, 512};
    const float* x0[3] = {x_c, x_m, x_d};

    for (int L = 0; L < 3; ++L) {
        const float* xc[3];
        float* xo[3];
        int inbuf  = (L == 0) ? 0 : (L - 1) & 1;   // L1 reads buf0, L2 reads buf1
        int outbuf = L & 1;                        // L0->0, L1->1, L2->0
        for (int t = 0; t < 3; ++t) {
            xc[t] = (L == 0) ? x0[t] : hbuf[inbuf][t];
            xo[t] = hbuf[outbuf][t];
        }
        for (int t = 0; t < 3; ++t) {
            const int n = nn[t], K = Kt[t], nr = nrel[t];
            hipMemsetAsync(Xaug, 0, (size_t)n * K * sizeof(float), stream);
            hipMemsetAsync(cnt,  0, (size_t)nr * n * sizeof(float), stream);
            for (int s = 0; s < nr; ++s) {
                int r = rel_of[t][s];
                int nedge = ne[r];
                int blocks = (nedge + 7) / 8;      // 8 waves/block, 1 edge/wave
                scatter_accum_kernel<<<blocks, 256, 0, stream>>>(
                    xc[srct_of[t][s]], E[r], E[r] + nedge,
                    Xaug, cnt + (size_t)s * n, nedge, K, s * DIM);
            }
            finalize_xaug_kernel<<<(n * DIM + 255) / 256, 256, 0, stream>>>(
                Xaug, cnt, xc[t], n, K, nr);
            build_wcomb_kernel<<<(128 * 128 + 255) / 256, 256, 0, stream>>>(
                Wl, Wr, bl, Wcomb, bsum, L,
                rel_of[t][0], rel_of[t][1], (nr > 2) ? rel_of[t][2] : 0, nr, K);
            gemm_wmma_kernel<<<(n + GEMM_ROWS - 1) / GEMM_ROWS, 256, 0, stream>>>(
                Xaug, Wcomb, bsum, xo[t], n, K, 1.0f / (float)nr, (L < 2) ? 1 : 0);
        }
    }

    // Final row-wise L2 norm into concatenated d_out (c, m, d).
    float* outp = (float*)d_out;
    size_t oo = 0;
    for (int t = 0; t < 3; ++t) {
        int n = nn[t];
        l2norm_kernel<<<(n + 7) / 8, 256, 0, stream>>>(hbuf[0][t], outp + oo, n);
        oo += (size_t)n * DIM;
    }
    (void)n_in; (void)out_size; (void)ws_size;
}